// TensorProduct_90185723281588
// MI455X (gfx1250) — compile-verified
//
#include <hip/hip_runtime.h>

// Clebsch-Gordan tensor product "uvu":
//   x1: [N,512] = 128x0e | 128x1o(3)     x2: [N,4] = 0e | 1o(3)
//   w : [N,5,128]                        out: [N,1408] = 128 | 384 | 384 | 128 | 384
// Memory-bound (~0.34 flop/byte): optimize purely for coalesced b128 traffic +
// CDNA5 async global->LDS double buffering. WMMA is inapplicable (no GEMM structure).

#define CG_MUL     128
#define ROW_X1     512
#define ROW_W      640
#define ROW_OUT    1408
#define WAVES      4
#define BLOCK      (WAVES * 32)
#define BUF_FLOATS (ROW_X1 + ROW_W)   // 1152 floats = 4608 B per staged row

#define INV_SQRT3f 0.57735026918962576451f
#define INV_SQRT2f 0.70710678118654752440f

typedef float v4 __attribute__((ext_vector_type(4)));
typedef int   i4 __attribute__((ext_vector_type(4)));

// Builtin signature (leaked by clang diagnostic): src is int4 in AS(1).
typedef __attribute__((address_space(1))) i4 gas_i4;   // global int4
typedef __attribute__((address_space(3))) i4 las_i4;   // LDS int4

#if defined(__HIP_DEVICE_COMPILE__) &&                                  \
    __has_builtin(__builtin_amdgcn_global_load_async_to_lds_b128) &&    \
    __has_builtin(__builtin_amdgcn_s_wait_asynccnt)
#define USE_ASYNC_LDS 1
#endif

__device__ __forceinline__ v4 ld4(const float* p) { return *(const v4*)p; }

// Per-lane compute for 4 channels (c0..c0+3) of one row, then b128 stores.
__device__ __forceinline__ void compute_store(
    v4 s, v4 a0, v4 a1, v4 a2,
    v4 W0, v4 W1, v4 W2, v4 W3, v4 W4,
    v4 xv, float* __restrict__ outr, int c0)
{
    const float b0 = xv[0], b1 = xv[1], b2 = xv[2], b3 = xv[3];
    float av[12] = {a0[0],a0[1],a0[2],a0[3],
                    a1[0],a1[1],a1[2],a1[3],
                    a2[0],a2[1],a2[2],a2[3]};
    v4 o0, o3;
    float o1[12], o2[12], o4[12];
#pragma unroll
    for (int k = 0; k < 4; ++k) {
        const float x0 = s[k];
        const float v0 = av[3*k+0], v1 = av[3*k+1], v2 = av[3*k+2];
        // out0 = w0 * x1_0 * x2_0
        o0[k] = W0[k] * x0 * b0;
        // out1 = w1 * x1_0 (outer) x2_1
        o1[3*k+0] = W1[k] * x0 * b1;
        o1[3*k+1] = W1[k] * x0 * b2;
        o1[3*k+2] = W1[k] * x0 * b3;
        // out2 = w2 * x1_1 * x2_0
        o2[3*k+0] = W2[k] * v0 * b0;
        o2[3*k+1] = W2[k] * v1 * b0;
        o2[3*k+2] = W2[k] * v2 * b0;
        // out3 = w3 * (x1_1 . x2_1) / sqrt(3)
        const float dt = v0*b1 + v1*b2 + v2*b3;
        o3[k] = W3[k] * dt * INV_SQRT3f;
        // out4 = w4 * (x1_1 x x2_1) / sqrt(2)
        const float cx = v1*b3 - v2*b2;
        const float cy = v2*b1 - v0*b3;
        const float cz = v0*b2 - v1*b1;
        o4[3*k+0] = W4[k] * cx * INV_SQRT2f;
        o4[3*k+1] = W4[k] * cy * INV_SQRT2f;
        o4[3*k+2] = W4[k] * cz * INV_SQRT2f;
    }
    const int t3 = 3 * c0;
    *(v4*)(outr + c0) = o0;
    { v4 t = {o1[0],o1[1],o1[2],o1[3]};  *(v4*)(outr + 128 + t3)      = t; }
    { v4 t = {o1[4],o1[5],o1[6],o1[7]};  *(v4*)(outr + 128 + t3 + 4)  = t; }
    { v4 t = {o1[8],o1[9],o1[10],o1[11]};*(v4*)(outr + 128 + t3 + 8)  = t; }
    { v4 t = {o2[0],o2[1],o2[2],o2[3]};  *(v4*)(outr + 512 + t3)      = t; }
    { v4 t = {o2[4],o2[5],o2[6],o2[7]};  *(v4*)(outr + 512 + t3 + 4)  = t; }
    { v4 t = {o2[8],o2[9],o2[10],o2[11]};*(v4*)(outr + 512 + t3 + 8)  = t; }
    *(v4*)(outr + 896 + c0) = o3;
    { v4 t = {o4[0],o4[1],o4[2],o4[3]};  *(v4*)(outr + 1024 + t3)     = t; }
    { v4 t = {o4[4],o4[5],o4[6],o4[7]};  *(v4*)(outr + 1024 + t3 + 4) = t; }
    { v4 t = {o4[8],o4[9],o4[10],o4[11]};*(v4*)(outr + 1024 + t3 + 8) = t; }
}

__global__ __launch_bounds__(BLOCK) void cg_tp_kernel(
    const float* __restrict__ x1, const float* __restrict__ x2,
    const float* __restrict__ w, float* __restrict__ out, int N)
{
    const int lane   = (int)(threadIdx.x & 31u);
    const int wv     = __builtin_amdgcn_readfirstlane((int)(threadIdx.x >> 5));
    const int waveId = (int)blockIdx.x * WAVES + wv;
    const int stride = (int)gridDim.x * WAVES;
    const int c0     = lane * 4;

#ifdef USE_ASYNC_LDS
    // CDNA5 async global->LDS double buffering: 9 b128 async ops stage one row
    // (x1: 4 ops, w: 5 ops); s_wait_asynccnt gates consumption. Per-wave
    // private buffers -> no barriers.
    __shared__ float lds[WAVES][2][BUF_FLOATS];

    auto stage = [&](int r, int b) {
        const float* sx = x1 + (size_t)r * ROW_X1 + c0;
        const float* sw = w  + (size_t)r * ROW_W  + c0;
        float* d = &lds[wv][b][c0];
#pragma unroll
        for (int i = 0; i < 4; ++i)
            __builtin_amdgcn_global_load_async_to_lds_b128(
                (gas_i4*)(unsigned long long)(sx + i * CG_MUL),
                (las_i4*)(unsigned)(unsigned long long)(d + i * CG_MUL),
                0, 0);
#pragma unroll
        for (int i = 0; i < 5; ++i)
            __builtin_amdgcn_global_load_async_to_lds_b128(
                (gas_i4*)(unsigned long long)(sw + i * CG_MUL),
                (las_i4*)(unsigned)(unsigned long long)(d + ROW_X1 + i * CG_MUL),
                0, 0);
    };

    int r = waveId;
    int b = 0;
    if (r < N) stage(r, 0);
    while (r < N) {
        const int rn = r + stride;
        if (rn < N) {
            stage(rn, b ^ 1);
            __builtin_amdgcn_s_wait_asynccnt(9);   // current buffer's 9 ops done
        } else {
            __builtin_amdgcn_s_wait_asynccnt(0);
        }
        const float* base = &lds[wv][b][0];
        const int t3 = 3 * c0;
        v4 s  = ld4(base + c0);
        v4 a0 = ld4(base + CG_MUL + t3);
        v4 a1 = ld4(base + CG_MUL + t3 + 4);
        v4 a2 = ld4(base + CG_MUL + t3 + 8);
        v4 W0 = ld4(base + ROW_X1 + 0 * CG_MUL + c0);
        v4 W1 = ld4(base + ROW_X1 + 1 * CG_MUL + c0);
        v4 W2 = ld4(base + ROW_X1 + 2 * CG_MUL + c0);
        v4 W3 = ld4(base + ROW_X1 + 3 * CG_MUL + c0);
        v4 W4 = ld4(base + ROW_X1 + 4 * CG_MUL + c0);
        v4 xv = ld4(x2 + (size_t)r * 4);           // wave-uniform -> SMEM load
        compute_store(s, a0, a1, a2, W0, W1, W2, W3, W4, xv,
                      out + (size_t)r * ROW_OUT, c0);
        b ^= 1;
        r = rn;
    }
#else
    // Fallback: direct coalesced b128 loads + prefetch of the next row.
    for (int r = waveId; r < N; r += stride) {
        const float* sx = x1 + (size_t)r * ROW_X1;
        const float* sw = w  + (size_t)r * ROW_W;
        const int rn = r + stride;
        if (rn < N) {
            __builtin_prefetch(x1 + (size_t)rn * ROW_X1 + c0, 0, 0);
            __builtin_prefetch(w  + (size_t)rn * ROW_W  + c0, 0, 0);
        }
        const int t3 = 3 * c0;
        v4 s  = ld4(sx + c0);
        v4 a0 = ld4(sx + CG_MUL + t3);
        v4 a1 = ld4(sx + CG_MUL + t3 + 4);
        v4 a2 = ld4(sx + CG_MUL + t3 + 8);
        v4 W0 = ld4(sw + 0 * CG_MUL + c0);
        v4 W1 = ld4(sw + 1 * CG_MUL + c0);
        v4 W2 = ld4(sw + 2 * CG_MUL + c0);
        v4 W3 = ld4(sw + 3 * CG_MUL + c0);
        v4 W4 = ld4(sw + 4 * CG_MUL + c0);
        v4 xv = ld4(x2 + (size_t)r * 4);
        compute_store(s, a0, a1, a2, W0, W1, W2, W3, W4, xv,
                      out + (size_t)r * ROW_OUT, c0);
    }
#endif
}

extern "C" void kernel_launch(void* const* d_in, const int* in_sizes, int n_in,
                              void* d_out, int out_size, void* d_ws, size_t ws_size,
                              hipStream_t stream) {
    (void)n_in; (void)out_size; (void)d_ws; (void)ws_size;
    const float* x1 = (const float*)d_in[0];
    const float* x2 = (const float*)d_in[1];
    const float* w  = (const float*)d_in[2];
    float* out = (float*)d_out;
    const int N = in_sizes[0] / ROW_X1;          // 50000
    // ~4 rows per wave so the async pipeline has iterations to overlap.
    int waves  = (N + 3) / 4;
    int blocks = (waves + WAVES - 1) / WAVES;
    if (blocks < 1) blocks = 1;
    cg_tp_kernel<<<blocks, BLOCK, 0, stream>>>(x1, x2, w, out, N);
}